// TopologicalLoss22_61984968016098
// MI455X (gfx1250) — compile-verified
//
#include <hip/hip_runtime.h>
#include <cstdint>

// ---------------------------------------------------------------------------
// TopologicalLoss: 3x3 "S minus one neighbor" stencil + 8 sigmoids + reduce.
// Roofline: 64MB fp32 read (~2.9us @ 23.3TB/s) vs 134M sigmoids -> the
// transcendental pipe is the ceiling, so: v_tanh_f32 sigmoid (1 trans/channel),
// async global->LDS halo staging (ASYNCcnt), wave32 shuffle reduction, and a
// single V_WMMA_F32_16X16X4_F32 ones-matrix reduction for the final scalar.
// ---------------------------------------------------------------------------

#define IMG    4096
#define TILE   64
#define HALO   66           // TILE + 2
#define LDS_W  68           // padded row stride (floats) -> conflict-free
#define BLOCK  256
#define ROWS_PER_THREAD 16  // TILE*TILE / BLOCK pixels per thread, 1-wide col

// ---- feature probes --------------------------------------------------------
#if defined(__has_builtin)
#  if __has_builtin(__builtin_amdgcn_global_load_async_to_lds_b32)
#    define USE_ASYNC_LDS 1
#  endif
#  if __has_builtin(__builtin_amdgcn_s_wait_asynccnt)
#    define HAVE_WAIT_ASYNC_BUILTIN 1
#  endif
#  if __has_builtin(__builtin_amdgcn_tanhf)
#    define HW_TANH(x) __builtin_amdgcn_tanhf(x)
#  elif __has_builtin(__builtin_amdgcn_tanh_f32)
#    define HW_TANH(x) __builtin_amdgcn_tanh_f32(x)
#  endif
#  if __has_builtin(__builtin_amdgcn_wmma_f32_16x16x4_f32)
#    define USE_WMMA_REDUCE 1
#  endif
#endif

typedef __attribute__((address_space(1))) int global_i32;
typedef __attribute__((address_space(3))) int lds_i32;
typedef float v2f __attribute__((ext_vector_type(2)));
typedef float v8f __attribute__((ext_vector_type(8)));

__device__ __forceinline__ float fast_sigmoid(float x) {
#ifdef HW_TANH
  // sigmoid(x) = 0.5 + 0.5*tanh(x/2)  -> single v_tanh_f32 (gfx1250 trans op)
  return fmaf(0.5f, HW_TANH(0.5f * x), 0.5f);
#else
  // fallback: 1/(1+exp(-x)) via hardware exp2 + rcp
  return __builtin_amdgcn_rcpf(
      1.0f + __builtin_amdgcn_exp2f(-1.4426950408889634f * x));
#endif
}

__device__ __forceinline__ void stage_to_lds(const float* __restrict__ gsrc,
                                             float* ldst) {
#ifdef USE_ASYNC_LDS
  __builtin_amdgcn_global_load_async_to_lds_b32(
      (global_i32*)gsrc,      // C-style cast: const_cast + addrspacecast
      (lds_i32*)ldst,
      /*imm offset*/ 0, /*cpol*/ 0);
#else
  *ldst = *gsrc;
#endif
}

__device__ __forceinline__ void wait_async_stage() {
#ifdef USE_ASYNC_LDS
#  ifdef HAVE_WAIT_ASYNC_BUILTIN
  __builtin_amdgcn_s_wait_asynccnt(0);
#  else
  asm volatile("s_wait_asynccnt 0" ::: "memory");
#  endif
#endif
}

// ---------------------------------------------------------------------------
// Stage 1: one 64x64 tile per block; partial sum per block into d_ws.
// ---------------------------------------------------------------------------
__global__ __launch_bounds__(BLOCK)
void topo_tile_kernel(const float* __restrict__ mask,
                      float* __restrict__ partial) {
  __shared__ float sm[HALO * LDS_W];
  __shared__ float warpsum[BLOCK / 32];

  const int tid = threadIdx.x;
  const int bx  = blockIdx.x, by = blockIdx.y;
  const int gx0 = bx * TILE - 1;       // halo origin in image coords
  const int gy0 = by * TILE - 1;

  // ---- stage 66x66 halo tile into LDS; OOB neighbors read as 1.0 ----------
  for (int idx = tid; idx < HALO * HALO; idx += BLOCK) {
    const int r  = idx / HALO;
    const int c  = idx - r * HALO;
    const int gy = gy0 + r;
    const int gx = gx0 + c;
    float* dst   = &sm[r * LDS_W + c];
    if ((unsigned)gy < (unsigned)IMG && (unsigned)gx < (unsigned)IMG) {
      stage_to_lds(mask + (size_t)gy * IMG + gx, dst);
    } else {
      *dst = 1.0f;   // pad value
    }
  }
  wait_async_stage();
  __syncthreads();

  // ---- each thread: 1-wide column, 16 rows, sliding 3x3 register window ---
  const int x  = tid & (TILE - 1);          // local column (lanes -> coalesced LDS)
  const int y0 = (tid >> 6) * ROWS_PER_THREAD;

  float a0 = sm[(y0 + 0) * LDS_W + x];      // top row of window
  float a1 = sm[(y0 + 0) * LDS_W + x + 1];
  float a2 = sm[(y0 + 0) * LDS_W + x + 2];
  float b0 = sm[(y0 + 1) * LDS_W + x];      // middle row (b1 = center pixel)
  float b1 = sm[(y0 + 1) * LDS_W + x + 1];
  float b2 = sm[(y0 + 1) * LDS_W + x + 2];

  float sa = (a0 + a1) + a2;                // sliding row-triple sums
  float sb = (b0 + b1) + b2;

  float acc = 0.0f;
#pragma unroll
  for (int i = 0; i < ROWS_PER_THREAD; ++i) {
    const int rc = (y0 + 2 + i) * LDS_W + x;
    const float c0 = sm[rc];
    const float c1 = sm[rc + 1];
    const float c2 = sm[rc + 2];
    const float sc = (c0 + c1) + c2;

    // 8-neighbor sum (kernel center is always 0): S = sa + (b0+b2) + sc
    const float S = (sa + (sb - b1)) + sc;
    const float B = fmaf(-10.0f, S, 10.0f);          // 10*(1 - S)

    // channel 7: full sum; channels 0..6: each excludes exactly one of
    // {NW,N,NE,E,SW,S,SE} (west b0 is never excluded) -> nc = S - n
    float s = fast_sigmoid(B);
    s += fast_sigmoid(fmaf(10.0f, a0, B));
    s += fast_sigmoid(fmaf(10.0f, a1, B));
    s += fast_sigmoid(fmaf(10.0f, a2, B));
    s += fast_sigmoid(fmaf(10.0f, b2, B));
    s += fast_sigmoid(fmaf(10.0f, c0, B));
    s += fast_sigmoid(fmaf(10.0f, c1, B));
    s += fast_sigmoid(fmaf(10.0f, c2, B));

    acc = fmaf(b1, s, acc);                          // * pm center

    a0 = b0; a1 = b1; a2 = b2;                       // slide window down
    b0 = c0; b1 = c1; b2 = c2;
    sa = sb; sb = sc;
  }

  // ---- wave32 + cross-wave reduction --------------------------------------
#pragma unroll
  for (int off = 16; off > 0; off >>= 1)
    acc += __shfl_xor(acc, off, 32);
  if ((tid & 31) == 0) warpsum[tid >> 5] = acc;
  __syncthreads();
  if (tid == 0) {
    float t = 0.0f;
#pragma unroll
    for (int w = 0; w < BLOCK / 32; ++w) t += warpsum[w];
    partial[by * (IMG / TILE) + bx] = t;
  }
}

// ---------------------------------------------------------------------------
// Stage 2: deterministic reduction of 4096 block partials -> scalar loss.
// Final 64-value sum is done on the matrix unit: D = A(16x4) x ones(4x16),
// so D[m][n] = rowsum(m) for every n; lane0 holds M=0..7 of column 0 and
// lane16 holds M=8..15, hence total = t + shfl_xor(t,16).
// ---------------------------------------------------------------------------
__global__ __launch_bounds__(BLOCK)
void topo_reduce_kernel(const float* __restrict__ partial,
                        float* __restrict__ out) {
  __shared__ float red[BLOCK];
  const int tid = threadIdx.x;
  const int n = (IMG / TILE) * (IMG / TILE);   // 4096
  float s = 0.0f;
  for (int i = tid; i < n; i += BLOCK) s += partial[i];
  red[tid] = s;
  __syncthreads();
  if (tid < 64)
    red[tid] = (red[tid] + red[tid + 64]) + (red[tid + 128] + red[tid + 192]);
  __syncthreads();

#ifdef USE_WMMA_REDUCE
  // All 256 threads execute the WMMA (EXEC must be all-ones in every wave).
  // A-matrix 16x4 f32 layout: lanes 0-15 hold {K0,K1} of row M=lane,
  // lanes 16-31 hold {K2,K3} of row M=lane-16 -> logical index 4*M+K.
  const int lane = tid & 31;
  const int idx0 = 4 * (lane & 15) + ((lane >> 4) << 1);
  v2f a; a.x = red[idx0]; a.y = red[idx0 + 1];
  v2f b; b.x = 1.0f; b.y = 1.0f;               // B = ones(4x16)
  v8f c = {};
  v8f d = __builtin_amdgcn_wmma_f32_16x16x4_f32(
      /*neg_a=*/false, a, /*neg_b=*/false, b,
      /*c_mod=*/(short)0, c, /*reuse_a=*/false, /*reuse_b=*/false);
  float t = ((d[0] + d[1]) + (d[2] + d[3])) + ((d[4] + d[5]) + (d[6] + d[7]));
  t += __shfl_xor(t, 16, 32);                  // M=0..7 half + M=8..15 half
  if (tid == 0) out[0] = t * 0.125f;           // the /8 channel average
#else
  if (tid == 0) {
    float t = 0.0f;
#pragma unroll
    for (int i = 0; i < 64; ++i) t += red[i];
    out[0] = t * 0.125f;
  }
#endif
}

// ---------------------------------------------------------------------------
extern "C" void kernel_launch(void* const* d_in, const int* in_sizes, int n_in,
                              void* d_out, int out_size, void* d_ws,
                              size_t ws_size, hipStream_t stream) {
  (void)in_sizes; (void)n_in; (void)out_size; (void)ws_size;
  const float* mask = (const float*)d_in[0];
  // d_in[1] (kernels) unused: the 8 kernels are hardcoded as "S minus one
  // neighbor" per the reference's constant _KERNELS table.
  float* partial = (float*)d_ws;               // 4096 floats of scratch

  dim3 grid(IMG / TILE, IMG / TILE);
  topo_tile_kernel<<<grid, BLOCK, 0, stream>>>(mask, partial);
  topo_reduce_kernel<<<1, BLOCK, 0, stream>>>(partial, (float*)d_out);
}